// Spatial_Positional_Encoding_85564338471083
// MI455X (gfx1250) — compile-verified
//
#include <hip/hip_runtime.h>

// out[b, c, d] = x[b,c,d] + (x[b,c,d] - x[b,0,d]),  x: (B, 32, 512) f32
// Memory-bound: ~520 MiB of HBM traffic -> ~22 us at 23.3 TB/s.
// Strategy: float4 (b128) coalesced streaming; central row staged to LDS once
// per block via gfx1250 async global->LDS copy (ASYNCcnt path).

typedef float float4v __attribute__((ext_vector_type(4)));

#define ROWS   32
#define COLS   512   // floats per row
#define COLS4  128   // float4 per row (2 KB per central row)

__global__ __launch_bounds__(256) void spe_bcast_kernel(
    const float* __restrict__ x, float* __restrict__ out)
{
    const int b    = blockIdx.x;
    const int t    = threadIdx.x;
    const int col4 = t & (COLS4 - 1);   // fixed column of this thread
    const int half = t >> 7;            // 0 or 1: which row of the pair

    const float4v* __restrict__ xb = (const float4v*)(x   + (size_t)b * ROWS * COLS);
    float4v* __restrict__       ob = (float4v*)      (out + (size_t)b * ROWS * COLS);

    // Central row cache in LDS.
    __shared__ float4v cent[COLS4];

    // Stage row 0 (2048 B) into LDS with the CDNA5 async global->LDS copy.
    // Threads 0..127 (waves 0..3, full EXEC per wave) each move 16 bytes.
    //
    // LDS address: per the flat-aperture rules, the low 32 bits of a generic
    // pointer into LDS are the LDS byte offset. Taking the address via
    // ptrtoint also makes `cent` escape, so the "memory"-clobbering asm is
    // known to write it and the later ds_load cannot be folded to poison.
    if (t < COLS4) {
        unsigned long long gaddr   = (unsigned long long)(xb + col4);
        unsigned int       ldsaddr = (unsigned int)(uintptr_t)&cent[col4];
        asm volatile("global_load_async_to_lds_b128 %0, %1, off"
                     :
                     : "v"(ldsaddr), "v"(gaddr)
                     : "memory");
    }
    asm volatile("s_wait_asynccnt 0" ::: "memory");
    __syncthreads();

    // Loop-invariant broadcast value lives in registers for all iterations.
    const float4v c = cent[col4];

    // 32 rows, two rows per iteration (128 threads each), b128 load/store.
    #pragma unroll
    for (int it = 0; it < ROWS / 2; ++it) {
        const int row = it * 2 + half;
        const int idx = row * COLS4 + col4;
        const float4v v = xb[idx];
        ob[idx] = v + (v - c);   // match reference association x + (x - central)
    }
}

extern "C" void kernel_launch(void* const* d_in, const int* in_sizes, int n_in,
                              void* d_out, int out_size, void* d_ws, size_t ws_size,
                              hipStream_t stream)
{
    const float* x  = (const float*)d_in[0];
    float*      out = (float*)d_out;

    const int B = in_sizes[0] / (ROWS * COLS);   // 4096 for the reference shapes

    spe_bcast_kernel<<<B, 256, 0, stream>>>(x, out);
}